// Net_47811575939419
// MI455X (gfx1250) — compile-verified
//
#include <hip/hip_runtime.h>
#include <hip/hip_bf16.h>

typedef __attribute__((ext_vector_type(2))) float v2f;
typedef __attribute__((ext_vector_type(8))) float v8f;

__device__ __forceinline__ void atomicAddF(float* p, float v) {
    __hip_atomic_fetch_add(p, v, __ATOMIC_RELAXED, __HIP_MEMORY_SCOPE_AGENT);
}

// -------------------- degree / symmetric norm --------------------
__global__ void k_init_deg(float* __restrict__ deg, int n) {
    int i = blockIdx.x * blockDim.x + threadIdx.x;
    if (i < n) deg[i] = 1.0f;               // self-loop
}
__global__ void k_deg_edges(const int* __restrict__ col, float* __restrict__ deg, int e) {
    int i = blockIdx.x * blockDim.x + threadIdx.x;
    if (i < e) atomicAddF(&deg[col[i]], 1.0f);
}
__global__ void k_dis(float* __restrict__ deg, int n) {
    int i = blockIdx.x * blockDim.x + threadIdx.x;
    if (i < n) deg[i] = rsqrtf(deg[i]);     // deg >= 1 guaranteed
}

// -------------------- generic fp32 WMMA GEMM: C[n,O] = A[n,K] @ W[O,K]^T (+bias) --------------------
// One wave computes a 16-row strip x OT 16-col tiles of C. K is compile-time: the k-loop fully
// unrolls so every load has a constant immediate offset from a loop-invariant base pointer,
// letting the scheduler batch loads deep ahead of the wmma chain.
// A-frag (16x4 f32): lanes 0-15 -> rows M=0..15, v0/v1 = K=k,k+1; lanes 16-31 -> K=k+2,k+3.
// B-frag (4x16):     lanes index N; lane-half selects K pair (mirror of A).
// C:                 lane holds column N=l16; VGPR v -> row M = half*8 + v.
template <int K, int OT, bool GUARD>
__global__ __launch_bounds__(128) void k_gemm_wmma(
        const float* __restrict__ A, const float* __restrict__ W,
        const float* __restrict__ bias, float* __restrict__ C,
        int n, int O, int hasBias) {
    int wid  = threadIdx.x >> 5;
    int lane = threadIdx.x & 31;
    int half = lane >> 4;
    int l16  = lane & 15;

    int tilesM = n >> 4;
    int tilesO = (O + 15) >> 4;
    int nOG    = (tilesO + OT - 1) / OT;     // o-tile groups per row strip
    int tile   = blockIdx.x * 4 + wid;
    int total  = tilesM * nOG;
    if (tile >= total) return;               // wave-uniform

    int mt = tile % tilesM;                  // consecutive waves share the o-group -> W stays hot
    int og = tile / tilesM;
    int m0 = mt << 4;
    int obase = og * (OT * 16);

    const float* ap = A + (size_t)(m0 + l16) * K + 2 * half;

    const float* wp[OT];
    float msk[OT];
    int oc[OT];
#pragma unroll
    for (int t = 0; t < OT; ++t) {
        int o = obase + t * 16 + l16;
        oc[t] = o;
        int ocl = GUARD ? ((o < O) ? o : (O - 1)) : o;
        msk[t] = (!GUARD || o < O) ? 1.0f : 0.0f;
        wp[t] = W + (size_t)ocl * K + 2 * half;
    }

    v8f acc[OT];
#pragma unroll
    for (int t = 0; t < OT; ++t) acc[t] = (v8f){};

#pragma unroll
    for (int k = 0; k < K; k += 4) {
        v2f a;
        a.x = ap[k];
        a.y = ap[k + 1];
#pragma unroll
        for (int t = 0; t < OT; ++t) {
            v2f b;
            if (GUARD) {
                b.x = wp[t][k] * msk[t];
                b.y = wp[t][k + 1] * msk[t];
            } else {
                b.x = wp[t][k];
                b.y = wp[t][k + 1];
            }
            acc[t] = __builtin_amdgcn_wmma_f32_16x16x4_f32(
                    /*neg_a=*/false, a, /*neg_b=*/false, b,
                    /*c_mod=*/(short)0, acc[t], /*reuse_a=*/false, /*reuse_b=*/false);
        }
    }

#pragma unroll
    for (int t = 0; t < OT; ++t) {
        int o = oc[t];
        if (!GUARD || o < O) {
            float bv = hasBias ? bias[o] : 0.0f;
            float* cp = C + o;
#pragma unroll
            for (int vi = 0; vi < 8; ++vi) {
                int crow = m0 + half * 8 + vi;
                cp[(size_t)crow * O] = acc[t][vi] + bv;
            }
        }
    }
}

// -------------------- GCN aggregation: H[c] += dis[r]*dis[c]*T[r]; self-loop; +bias; relu ------------
__global__ void k_agg_self(const float* __restrict__ T, const float* __restrict__ dis,
                           float* __restrict__ H, int n, int oshift) {
    long idx = (long)blockIdx.x * blockDim.x + threadIdx.x;
    if (idx >= ((long)n << oshift)) return;
    int i = (int)(idx >> oshift);
    float d = dis[i];
    H[idx] = d * d * T[idx];
}
__global__ void k_agg_edges(const float* __restrict__ T, const float* __restrict__ dis,
                            const int* __restrict__ row, const int* __restrict__ col,
                            float* __restrict__ H, int e, int oshift, int omask) {
    long idx = (long)blockIdx.x * blockDim.x + threadIdx.x;
    if (idx >= ((long)e << oshift)) return;
    int ei = (int)(idx >> oshift);
    int f  = (int)(idx & omask);
    int r = row[ei], c = col[ei];
    float nrm = dis[r] * dis[c];
    atomicAddF(&H[((size_t)c << oshift) + f], nrm * T[((size_t)r << oshift) + f]);
}
__global__ void k_agg_finish(float* __restrict__ H, const float* __restrict__ b,
                             int n, int oshift, int omask) {
    long idx = (long)blockIdx.x * blockDim.x + threadIdx.x;
    if (idx >= ((long)n << oshift)) return;
    int f = (int)(idx & omask);
    float v = H[idx] + b[f];
    H[idx] = v > 0.0f ? v : 0.0f;
}

// -------------------- fused edge kernel: Pout = (H[row]+H[col]+M[e](+Pprev[e])) @ Wl^T + bl ----------
// One wave per 16-edge tile: gather-sum into LDS, then OT independent WMMA chains over the
// shared LDS A fragment. IN compile-time (power of two <= 128); OT*16 covers OUT.
template <int IN, int OT, bool GUARD>
__global__ __launch_bounds__(128) void k_edge_wmma(
        const float* __restrict__ H, const float* __restrict__ M,
        const float* __restrict__ Pprev, int hasP,
        const int* __restrict__ row, const int* __restrict__ col,
        const float* __restrict__ Wl, const float* __restrict__ bl,
        float* __restrict__ Pout, int E, int OUT) {
    constexpr int inShift = (IN == 128) ? 7 : (IN == 64) ? 6 : (IN == 32) ? 5 : (IN == 16) ? 4 : 3;
    __shared__ float smem[4 * 16 * 128];    // 32 KB: 4 waves x (16 edges x up to 128 feats)
    int wid  = threadIdx.x >> 5;
    int lane = threadIdx.x & 31;
    int half = lane >> 4;
    int l16  = lane & 15;

    float* sw = smem + wid * (16 * IN);
    int tile = blockIdx.x * 4 + wid;
    int e0   = tile << 4;
    bool active = (e0 < E);                 // wave-uniform

    if (active) {
#pragma unroll
        for (int it = 0; it < (16 * IN) / 32; ++it) {
            int idx = it * 32 + lane;
            int ei = idx >> inShift;
            int f  = idx & (IN - 1);
            int e  = e0 + ei;
            float v = 0.0f;
            if (e < E) {
                int r = row[e], c = col[e];
                v = H[((size_t)r << inShift) + f] + H[((size_t)c << inShift) + f]
                  + M[((size_t)e << inShift) + f];
                if (hasP) v += Pprev[((size_t)e << inShift) + f];
            }
            sw[idx] = v;
        }
    }
    __syncthreads();
    if (!active) return;                    // wave-uniform

    const float* wp[OT];
    float msk[OT];
    int oc[OT];
#pragma unroll
    for (int t = 0; t < OT; ++t) {
        int o = t * 16 + l16;
        oc[t] = o;
        int ocl = GUARD ? ((o < OUT) ? o : (OUT - 1)) : o;
        msk[t] = (!GUARD || o < OUT) ? 1.0f : 0.0f;
        wp[t] = Wl + (size_t)ocl * IN + 2 * half;
    }
    const float* apx = sw + l16 * IN + 2 * half;

    v8f acc[OT];
#pragma unroll
    for (int t = 0; t < OT; ++t) acc[t] = (v8f){};

#pragma unroll
    for (int k = 0; k < IN; k += 4) {
        v2f a;
        a.x = apx[k];
        a.y = apx[k + 1];
#pragma unroll
        for (int t = 0; t < OT; ++t) {
            v2f b;
            if (GUARD) {
                b.x = wp[t][k] * msk[t];
                b.y = wp[t][k + 1] * msk[t];
            } else {
                b.x = wp[t][k];
                b.y = wp[t][k + 1];
            }
            acc[t] = __builtin_amdgcn_wmma_f32_16x16x4_f32(
                    false, a, false, b, (short)0, acc[t], false, false);
        }
    }

#pragma unroll
    for (int t = 0; t < OT; ++t) {
        int o = oc[t];
        if (!GUARD || o < OUT) {
            float bv = bl[o];
#pragma unroll
            for (int vi = 0; vi < 8; ++vi) {
                int e = e0 + half * 8 + vi;
                if (e < E) Pout[(size_t)e * OUT + o] = acc[t][vi] + bv;
            }
        }
    }
}

// -------------------- final: out[e] = dot(H4[row]+H4[col]+P3[e]+M4[e], Wcl) + bcl --------------------
__global__ void k_final(const float* __restrict__ H4, const float* __restrict__ M4,
                        const float* __restrict__ P3,
                        const int* __restrict__ row, const int* __restrict__ col,
                        const float* __restrict__ Wcl, const float* __restrict__ bcl,
                        float* __restrict__ out, int E) {
    int e = blockIdx.x * blockDim.x + threadIdx.x;
    if (e >= E) return;
    int r = row[e], c = col[e];
    float acc = bcl[0];
#pragma unroll
    for (int j = 0; j < 8; ++j) {
        float s = H4[(size_t)r * 8 + j] + H4[(size_t)c * 8 + j]
                + P3[(size_t)e * 8 + j] + M4[(size_t)e * 8 + j];
        acc += s * Wcl[j];
    }
    out[e] = acc;
}

// ==================================================================================================
static inline int cdiv(long a, long b) { return (int)((a + b - 1) / b); }

extern "C" void kernel_launch(void* const* d_in, const int* in_sizes, int n_in,
                              void* d_out, int out_size, void* d_ws, size_t ws_size,
                              hipStream_t stream) {
    const float* x   = (const float*)d_in[0];
    const int*   eix = (const int*)d_in[1];
    const int N = in_sizes[0] / 256;
    const int E = in_sizes[1] / 2;
    const int* row = eix;
    const int* col = eix + E;

    const float* Wg1 = (const float*)d_in[2];  const float* bg1 = (const float*)d_in[3];
    const float* Wg2 = (const float*)d_in[4];  const float* bg2 = (const float*)d_in[5];
    const float* Wg3 = (const float*)d_in[6];  const float* bg3 = (const float*)d_in[7];
    const float* Wg4 = (const float*)d_in[8];  const float* bg4 = (const float*)d_in[9];
    const float* Wl1 = (const float*)d_in[10]; const float* bl1 = (const float*)d_in[11];
    const float* Wl2 = (const float*)d_in[12]; const float* bl2 = (const float*)d_in[13];
    const float* Wl3 = (const float*)d_in[14]; const float* bl3 = (const float*)d_in[15];
    const float* Wm1 = (const float*)d_in[16]; const float* bm1 = (const float*)d_in[17];
    const float* Wm2 = (const float*)d_in[18]; const float* bm2 = (const float*)d_in[19];
    const float* Wm3 = (const float*)d_in[20]; const float* bm3 = (const float*)d_in[21];
    const float* Wm4 = (const float*)d_in[22]; const float* bm4 = (const float*)d_in[23];
    const float* Wcl = (const float*)d_in[24]; const float* bcl = (const float*)d_in[25];

    // workspace layout (floats); T/M share one scratch S, P3 aliases P1 (both dead by then)
    float* ws = (float*)d_ws;
    size_t off = 0;
    float* dis = ws + off; off += (size_t)N;
    float* S   = ws + off; off += (size_t)N * 128;   // t_k / m_k scratch
    float* H1  = ws + off; off += (size_t)N * 128;
    float* H2  = ws + off; off += (size_t)N * 64;
    float* H3  = ws + off; off += (size_t)N * 16;
    float* H4  = ws + off; off += (size_t)N * 8;
    float* P1  = ws + off; off += (size_t)E * 64;
    float* P2  = ws + off; off += (size_t)E * 16;
    float* P3  = P1;                                  // alias (P1 dead after p2)
    float* out = (float*)d_out;

    // degree / norm
    k_init_deg <<<cdiv(N,256), 256, 0, stream>>>(dis, N);
    k_deg_edges<<<cdiv(E,256), 256, 0, stream>>>(col, dis, E);
    k_dis      <<<cdiv(N,256), 256, 0, stream>>>(dis, N);

    auto gemmBlocks = [&](int n, int O, int OT) {
        int tilesM = n >> 4;
        int nOG = cdiv((O + 15) >> 4, OT);
        return cdiv((long)tilesM * nOG, 4);
    };
    auto agg = [&](const float* T, float* H, const float* bg, int oshift) {
        int omask = (1 << oshift) - 1;
        k_agg_self  <<<cdiv((long)N << oshift,256), 256, 0, stream>>>(T, dis, H, N, oshift);
        k_agg_edges <<<cdiv((long)E << oshift,256), 256, 0, stream>>>(T, dis, row, col, H, E, oshift, omask);
        k_agg_finish<<<cdiv((long)N << oshift,256), 256, 0, stream>>>(H, bg, N, oshift, omask);
    };
    const int eb = cdiv((E + 15) >> 4, 4);   // edge-kernel blocks

    // ---- layer 1 ----
    k_gemm_wmma<256,4,false><<<gemmBlocks(N,128,4), 128, 0, stream>>>(x, Wg1, nullptr, S, N, 128, 0);
    agg(S, H1, bg1, 7);                               // h1 = relu(aggregate + bg1)
    k_gemm_wmma<256,4,false><<<gemmBlocks(E,128,4), 128, 0, stream>>>(x, Wm1, bm1, S, E, 128, 1);
    k_edge_wmma<128,4,false><<<eb, 128, 0, stream>>>(H1, S, nullptr, 0, row, col,
                                                     Wl1, bl1, P1, E, 64);      // p1
    // ---- layer 2 ----
    k_gemm_wmma<128,4,false><<<gemmBlocks(N,64,4), 128, 0, stream>>>(H1, Wg2, nullptr, S, N, 64, 0);
    agg(S, H2, bg2, 6);
    k_gemm_wmma<256,4,false><<<gemmBlocks(E,64,4), 128, 0, stream>>>(x, Wm2, bm2, S, E, 64, 1);
    k_edge_wmma<64,1,false><<<eb, 128, 0, stream>>>(H2, S, P1, 1, row, col,
                                                    Wl2, bl2, P2, E, 16);       // p2
    // ---- layer 3 ----
    k_gemm_wmma<64,1,false><<<gemmBlocks(N,16,1), 128, 0, stream>>>(H2, Wg3, nullptr, S, N, 16, 0);
    agg(S, H3, bg3, 4);
    k_gemm_wmma<256,1,false><<<gemmBlocks(E,16,1), 128, 0, stream>>>(x, Wm3, bm3, S, E, 16, 1);
    k_edge_wmma<16,1,true><<<eb, 128, 0, stream>>>(H3, S, P2, 1, row, col,
                                                   Wl3, bl3, P3, E, 8);         // p3 (OUT=8 padded)
    // ---- layer 4 + classifier ----
    k_gemm_wmma<16,1,true><<<gemmBlocks(N,8,1), 128, 0, stream>>>(H3, Wg4, nullptr, S, N, 8, 0);
    agg(S, H4, bg4, 3);
    k_gemm_wmma<256,1,true><<<gemmBlocks(E,8,1), 128, 0, stream>>>(x, Wm4, bm4, S, E, 8, 1);
    k_final<<<cdiv(E,256), 256, 0, stream>>>(H4, S, P3, row, col, Wcl, bcl, out, E);
}